// Seq2SeqReinforce_19172734009349
// MI455X (gfx1250) — compile-verified
//
#include <hip/hip_runtime.h>
#include <math.h>

#define VOCAB   50257
#define VP      50304          // vocab padded to multiple of 128
#define EDIM    256
#define HDIM    256
#define NL      2
#define CDIM    64
#define BATCH   4
#define TLEN    48
#define HC      320            // H + C
#define MASK_TOK 103
#define LN_EPS  1e-5f

typedef __attribute__((ext_vector_type(2))) float v2f;
typedef __attribute__((ext_vector_type(8))) float v8f;

// ---------------- WMMA helpers (f32 16x16x4, exact f32 path on CDNA5) -------

__device__ __forceinline__ v8f wmma4(v2f a, v2f b, v8f c) {
  // 8 args: (neg_a, A, neg_b, B, c_mod, C, reuse_a, reuse_b)
  return __builtin_amdgcn_wmma_f32_16x16x4_f32(false, a, false, b, (short)0, c,
                                               false, false);
}

__device__ __forceinline__ v8f zero8() {
  v8f z = {0.f, 0.f, 0.f, 0.f, 0.f, 0.f, 0.f, 0.f};
  return z;
}

// A fragment: 16x4 tile, rows valid only for r < BATCH (batch padded to 16).
// Lane layout (ISA 7.12.2): lanes 0-15 hold K={kk,kk+1}, lanes 16-31 K={kk+2,kk+3}.
__device__ __forceinline__ v2f load_a4(const float* base, int ld, int kk, int lane) {
  const int r = lane & 15;
  const int k = kk + ((lane >> 4) << 1);
  v2f a;
  if (r < BATCH) {
    a.x = base[r * ld + k];
    a.y = base[r * ld + k + 1];
  } else {
    a.x = 0.f; a.y = 0.f;
  }
  return a;
}

// B fragment: B[k][n] = W[n0+n][k] for row-major W[N][K] ("x @ W.T" form).
__device__ __forceinline__ v2f load_bw(const float* W, int ldk, int n0, int kk, int lane) {
  const int n = lane & 15;
  const int k = kk + ((lane >> 4) << 1);
  const float* p = W + (size_t)(n0 + n) * ldk + k;
  v2f b;
  b.x = p[0];
  b.y = p[1];
  return b;
}

__device__ __forceinline__ float sigf(float x) { return 1.f / (1.f + expf(-x)); }

// Ordered-uint encoding of float for atomicMax-based max reduction.
__device__ __forceinline__ unsigned ford(float f) {
  unsigned u = __float_as_uint(f);
  return (u & 0x80000000u) ? ~u : (u | 0x80000000u);
}
__device__ __forceinline__ float fdec(unsigned u) {
  unsigned v = (u & 0x80000000u) ? (u & 0x7FFFFFFFu) : ~u;
  return __uint_as_float(v);
}

// ---------------- Threefry-2x32 (JAX-style PRNG for Gumbel noise) -----------

__device__ __forceinline__ unsigned rotl32(unsigned x, unsigned n) {
  return (x << n) | (x >> (32u - n));
}

__device__ __forceinline__ void threefry2x32(unsigned k0, unsigned k1,
                                             unsigned c0, unsigned c1,
                                             unsigned& r0, unsigned& r1) {
  unsigned ks0 = k0, ks1 = k1, ks2 = k0 ^ k1 ^ 0x1BD11BDAu;
  unsigned x0 = c0 + ks0, x1 = c1 + ks1;
#define TFR(Ra, Rb, Rc, Rd)                                                    \
  x0 += x1; x1 = rotl32(x1, Ra); x1 ^= x0;                                     \
  x0 += x1; x1 = rotl32(x1, Rb); x1 ^= x0;                                     \
  x0 += x1; x1 = rotl32(x1, Rc); x1 ^= x0;                                     \
  x0 += x1; x1 = rotl32(x1, Rd); x1 ^= x0;
  TFR(13, 15, 26, 6)  x0 += ks1; x1 += ks2 + 1u;
  TFR(17, 29, 16, 24) x0 += ks2; x1 += ks0 + 2u;
  TFR(13, 15, 26, 6)  x0 += ks0; x1 += ks1 + 3u;
  TFR(17, 29, 16, 24) x0 += ks1; x1 += ks2 + 4u;
  TFR(13, 15, 26, 6)  x0 += ks2; x1 += ks0 + 5u;
#undef TFR
  r0 = x0; r1 = x1;
}

// ---------------- Shared LSTM stack (persistent single-workgroup) ----------
// Runs jcount positions through the 2-layer stack. blockDim.x == 256 (8 waves).
// x is time-major [pos][BATCH][HDIM]; h/c/z live in LDS.
// Tile-pairing: each wave owns 8 contiguous N-tiles, processed as 4 pairs with
// two independent accumulators -> A loaded once per k-step, 2 independent WMMAs.
__device__ void lstm_run(const float* __restrict__ xbuf, int jcount,
                         const float* __restrict__ Wih, const float* __restrict__ Whh,
                         const float* __restrict__ bih, const float* __restrict__ bhh,
                         float* hbuf, float* cbuf, float* zbuf) {
  const int tid  = threadIdx.x;
  const int lane = tid & 31;
  const int wave = tid >> 5;

  for (int j = 0; j < jcount; ++j) {
    for (int l = 0; l < NL; ++l) {
      const float* cur   = (l == 0) ? (xbuf + (size_t)j * BATCH * HDIM) : hbuf;
      const float* hprev = hbuf + l * BATCH * HDIM;
      const float* Wih_l = Wih + (size_t)l * 4 * HDIM * HDIM;
      const float* Whh_l = Whh + (size_t)l * 4 * HDIM * HDIM;

      // z[4][1024] = cur @ Wih.T + hprev @ Whh.T
      for (int tp = 0; tp < 8; tp += 2) {
        const int n0 = (wave * 8 + tp) * 16;
        v8f acc0 = zero8();
        v8f acc1 = zero8();
#pragma unroll 4
        for (int kk = 0; kk < HDIM; kk += 4) {
          v2f a  = load_a4(cur, HDIM, kk, lane);
          v2f b0 = load_bw(Wih_l, HDIM, n0, kk, lane);
          v2f b1 = load_bw(Wih_l, HDIM, n0 + 16, kk, lane);
          acc0 = wmma4(a, b0, acc0);
          acc1 = wmma4(a, b1, acc1);
        }
#pragma unroll 4
        for (int kk = 0; kk < HDIM; kk += 4) {
          v2f a  = load_a4(hprev, HDIM, kk, lane);
          v2f b0 = load_bw(Whh_l, HDIM, n0, kk, lane);
          v2f b1 = load_bw(Whh_l, HDIM, n0 + 16, kk, lane);
          acc0 = wmma4(a, b0, acc0);
          acc1 = wmma4(a, b1, acc1);
        }
        const int n = lane & 15, half = lane >> 4;
        if (half == 0) {
#pragma unroll
          for (int m = 0; m < BATCH; ++m) {
            zbuf[m * (4 * HDIM) + n0 + n]      = acc0[m];
            zbuf[m * (4 * HDIM) + n0 + 16 + n] = acc1[m];
          }
        }
      }
      __syncthreads();

      // Gates: tid == hidden index; PyTorch order i,f,g,o.
      {
        const int hix = tid;
        const float bi = bih[l * 4 * HDIM + hix]       + bhh[l * 4 * HDIM + hix];
        const float bf = bih[l * 4 * HDIM + 256 + hix] + bhh[l * 4 * HDIM + 256 + hix];
        const float bg = bih[l * 4 * HDIM + 512 + hix] + bhh[l * 4 * HDIM + 512 + hix];
        const float bo = bih[l * 4 * HDIM + 768 + hix] + bhh[l * 4 * HDIM + 768 + hix];
#pragma unroll
        for (int b = 0; b < BATCH; ++b) {
          const float zi = zbuf[b * 4 * HDIM + hix] + bi;
          const float zf = zbuf[b * 4 * HDIM + 256 + hix] + bf;
          const float zg = zbuf[b * 4 * HDIM + 512 + hix] + bg;
          const float zo = zbuf[b * 4 * HDIM + 768 + hix] + bo;
          const int   hcidx = (l * BATCH + b) * HDIM + hix;
          const float cn = sigf(zf) * cbuf[hcidx] + sigf(zi) * tanhf(zg);
          cbuf[hcidx] = cn;
          hbuf[hcidx] = sigf(zo) * tanhf(cn);
        }
      }
      __syncthreads();
    }
  }
}

// ---------------- Kernels ---------------------------------------------------

__global__ void init_kernel(const int* __restrict__ input_ids, int* toks_ws,
                            float* out_scores, int* out_tokens,
                            unsigned* red_max, float* red_sum,
                            unsigned long long* red_arg) {
  const int tid = threadIdx.x;
  for (int idx = tid; idx < BATCH * TLEN; idx += blockDim.x) {
    const int b = idx / TLEN, j = idx % TLEN;
    const int tok = (j == 0) ? input_ids[b * TLEN] : 0;
    toks_ws[idx]    = tok;
    out_tokens[idx] = tok;
    out_scores[idx] = 0.f;
  }
  if (tid < BATCH) {
    red_max[tid] = 0u;
    red_sum[tid] = 0.f;
    red_arg[tid] = 0ull;
  }
}

// cls_states[b][c] = labels[b] * W_cls[c] + b_cls[c]
__global__ void cls_kernel(const float* __restrict__ labels,
                           const float* __restrict__ Wc,
                           const float* __restrict__ bc, float* cls) {
  const int tid = threadIdx.x;
  const int b = tid / CDIM, c = tid % CDIM;
  cls[tid] = labels[b] * Wc[c] + bc[c];
}

// Encoder: xe[j][b][:] = relu(emb[ids[b][j]] @ W_e2e.T + b_e2e), 12 blocks x 16 rows.
__global__ void enc_embed_kernel(const int* __restrict__ ids,
                                 const float* __restrict__ emb,
                                 const float* __restrict__ W,
                                 const float* __restrict__ bias, float* xe) {
  __shared__ int toks[16];
  const int tid = threadIdx.x, lane = tid & 31, wave = tid >> 5;
  const int r0 = blockIdx.x * 16;
  if (tid < 16) {
    const int r = r0 + tid;
    const int j = r / BATCH, b = r % BATCH;
    toks[tid] = ids[b * TLEN + j];
  }
  __syncthreads();
  // one tile-pair per wave: columns [wave*32, wave*32+32)
  const int n0 = wave * 32;
  v8f acc0 = zero8();
  v8f acc1 = zero8();
#pragma unroll 4
  for (int kk = 0; kk < EDIM; kk += 4) {
    const int r = lane & 15;
    const int k = kk + ((lane >> 4) << 1);
    const float* ap = emb + (size_t)toks[r] * EDIM + k;
    v2f a; a.x = ap[0]; a.y = ap[1];
    v2f b0 = load_bw(W, EDIM, n0, kk, lane);
    v2f b1 = load_bw(W, EDIM, n0 + 16, kk, lane);
    acc0 = wmma4(a, b0, acc0);
    acc1 = wmma4(a, b1, acc1);
  }
  const int n = lane & 15, half = lane >> 4;
#pragma unroll
  for (int i = 0; i < 8; ++i) {
    const int m = i + 8 * half;
    const float v0 = acc0[i] + bias[n0 + n];
    const float v1 = acc1[i] + bias[n0 + 16 + n];
    xe[(size_t)(r0 + m) * HDIM + n0 + n]      = fmaxf(v0, 0.f);
    xe[(size_t)(r0 + m) * HDIM + n0 + 16 + n] = fmaxf(v1, 0.f);
  }
}

// Decoder embed for position t-1 only (rows 0..3).
__global__ void dec_embed_kernel(const int* __restrict__ toks_ws,
                                 const float* __restrict__ emb,
                                 const float* __restrict__ W,
                                 const float* __restrict__ bias, float* xd, int t) {
  __shared__ int toks[16];
  const int tid = threadIdx.x, lane = tid & 31, wave = tid >> 5;
  if (tid < 16) toks[tid] = (tid < BATCH) ? toks_ws[tid * TLEN + (t - 1)] : 0;
  __syncthreads();
  const int n0 = wave * 32;
  v8f acc0 = zero8();
  v8f acc1 = zero8();
#pragma unroll 4
  for (int kk = 0; kk < EDIM; kk += 4) {
    const int r = lane & 15;
    const int k = kk + ((lane >> 4) << 1);
    const float* ap = emb + (size_t)toks[r] * EDIM + k;
    v2f a; a.x = ap[0]; a.y = ap[1];
    v2f b0 = load_bw(W, EDIM, n0, kk, lane);
    v2f b1 = load_bw(W, EDIM, n0 + 16, kk, lane);
    acc0 = wmma4(a, b0, acc0);
    acc1 = wmma4(a, b1, acc1);
  }
  const int n = lane & 15, half = lane >> 4;
  if (half == 0) {
#pragma unroll
    for (int m = 0; m < BATCH; ++m) {
      const float v0 = acc0[m] + bias[n0 + n];
      const float v1 = acc1[m] + bias[n0 + 16 + n];
      xd[(size_t)((t - 1) * BATCH + m) * HDIM + n0 + n]      = fmaxf(v0, 0.f);
      xd[(size_t)((t - 1) * BATCH + m) * HDIM + n0 + 16 + n] = fmaxf(v1, 0.f);
    }
  }
}

// Encoder LSTM: whole sequence, zero initial state, final (h,c) -> ws.
__global__ void enc_lstm_kernel(const float* __restrict__ xe,
                                const float* __restrict__ Wih,
                                const float* __restrict__ Whh,
                                const float* __restrict__ bih,
                                const float* __restrict__ bhh,
                                float* h_ws, float* c_ws) {
  __shared__ float hbuf[NL * BATCH * HDIM];
  __shared__ float cbuf[NL * BATCH * HDIM];
  __shared__ float zbuf[BATCH * 4 * HDIM];
  for (int i = threadIdx.x; i < NL * BATCH * HDIM; i += blockDim.x) {
    hbuf[i] = 0.f; cbuf[i] = 0.f;
  }
  __syncthreads();
  lstm_run(xe, TLEN, Wih, Whh, bih, bhh, hbuf, cbuf, zbuf);
  for (int i = threadIdx.x; i < NL * BATCH * HDIM; i += blockDim.x) {
    h_ws[i] = hbuf[i]; c_ws[i] = cbuf[i];
  }
}

// Decoder step t: replay prefix (j < t) from carried state, then fc1+LN.
__global__ void dec_step_kernel(const float* __restrict__ xd, float* h_ws, float* c_ws,
                                const float* __restrict__ cls,
                                const float* __restrict__ Wih,
                                const float* __restrict__ Whh,
                                const float* __restrict__ bih,
                                const float* __restrict__ bhh,
                                const float* __restrict__ W_fc1,
                                const float* __restrict__ b_fc1,
                                const float* __restrict__ ln_g,
                                const float* __restrict__ ln_b,
                                float* z1_out, int t) {
  __shared__ float hbuf[NL * BATCH * HDIM];
  __shared__ float cbuf[NL * BATCH * HDIM];
  __shared__ float zbuf[BATCH * 4 * HDIM];
  __shared__ float zin[BATCH * HC];
  __shared__ float z1s[BATCH * HC];
  const int tid = threadIdx.x, lane = tid & 31, wave = tid >> 5;

  for (int i = tid; i < NL * BATCH * HDIM; i += blockDim.x) {
    hbuf[i] = h_ws[i]; cbuf[i] = c_ws[i];
  }
  __syncthreads();

  lstm_run(xd, t, Wih, Whh, bih, bhh, hbuf, cbuf, zbuf);

  // z = [h_layer1 | cls]  -> fc1 -> relu -> LayerNorm -> z1_out
  for (int idx = tid; idx < BATCH * HC; idx += blockDim.x) {
    const int b = idx / HC, i = idx % HC;
    zin[idx] = (i < HDIM) ? hbuf[(1 * BATCH + b) * HDIM + i]
                          : cls[b * CDIM + (i - HDIM)];
  }
  __syncthreads();

  // 20 N-tiles = 10 pairs over 8 waves
  for (int pr = wave; pr < 10; pr += 8) {
    const int n0 = pr * 32;
    v8f acc0 = zero8();
    v8f acc1 = zero8();
#pragma unroll 4
    for (int kk = 0; kk < HC; kk += 4) {
      v2f a  = load_a4(zin, HC, kk, lane);
      v2f b0 = load_bw(W_fc1, HC, n0, kk, lane);
      v2f b1 = load_bw(W_fc1, HC, n0 + 16, kk, lane);
      acc0 = wmma4(a, b0, acc0);
      acc1 = wmma4(a, b1, acc1);
    }
    const int n = lane & 15, half = lane >> 4;
    if (half == 0) {
#pragma unroll
      for (int m = 0; m < BATCH; ++m) {
        const float v0 = acc0[m] + b_fc1[n0 + n];
        const float v1 = acc1[m] + b_fc1[n0 + 16 + n];
        z1s[m * HC + n0 + n]      = fmaxf(v0, 0.f);
        z1s[m * HC + n0 + 16 + n] = fmaxf(v1, 0.f);
      }
    }
  }
  __syncthreads();

  if (wave < BATCH) {
    const int b = wave;
    float s = 0.f, ss = 0.f;
    for (int i = lane; i < HC; i += 32) {
      const float v = z1s[b * HC + i];
      s += v; ss += v * v;
    }
    for (int off = 16; off > 0; off >>= 1) {
      s  += __shfl_xor(s, off);
      ss += __shfl_xor(ss, off);
    }
    const float mu  = s * (1.f / HC);
    const float var = ss * (1.f / HC) - mu * mu;
    const float rs  = rsqrtf(var + LN_EPS);
    for (int i = lane; i < HC; i += 32) {
      const float v = z1s[b * HC + i];
      z1_out[b * HC + i] = (v - mu) * rs * ln_g[i] + ln_b[i];
    }
  }
  __syncthreads();
  for (int i = tid; i < NL * BATCH * HDIM; i += blockDim.x) {
    h_ws[i] = hbuf[i]; c_ws[i] = cbuf[i];
  }
}

// fc2 sweep: 128 columns per block (4 waves x tile-pair), logits + running max
// (for logsumexp) + Gumbel-argmax via packed 64-bit atomicMax.
__global__ void fc2_kernel(const float* __restrict__ z1,
                           const float* __restrict__ W2,
                           const float* __restrict__ b2,
                           float* logits, unsigned* red_max,
                           unsigned long long* red_arg, int t) {
  __shared__ float zz[BATCH * HC];
  const int tid = threadIdx.x, lane = tid & 31, wave = tid >> 5;
  for (int i = tid; i < BATCH * HC; i += blockDim.x) zz[i] = z1[i];
  __syncthreads();

  const int n0 = blockIdx.x * 128 + wave * 32;     // tile pair: n0, n0+16
  const int n = lane & 15, half = lane >> 4;
  const int v0 = n0 + n;
  const int v1 = n0 + 16 + n;
  const bool val0 = (half == 0) && (v0 < VOCAB);
  const bool val1 = (half == 0) && (v1 < VOCAB);
  const int vc0 = (v0 < VOCAB) ? v0 : (VOCAB - 1);
  const int vc1 = (v1 < VOCAB) ? v1 : (VOCAB - 1);
  const float* wrow0 = W2 + (size_t)vc0 * HC;
  const float* wrow1 = W2 + (size_t)vc1 * HC;
  __builtin_prefetch(wrow0, 0, 0);  // global_prefetch_b8: warm L2
  __builtin_prefetch(wrow1, 0, 0);

  v8f acc0 = zero8();
  v8f acc1 = zero8();
#pragma unroll 4
  for (int kk = 0; kk < HC; kk += 4) {
    v2f a = load_a4(zz, HC, kk, lane);
    const int ko = kk + (half << 1);
    v2f b0; b0.x = wrow0[ko]; b0.y = wrow0[ko + 1];
    v2f b1; b1.x = wrow1[ko]; b1.y = wrow1[ko + 1];
    acc0 = wmma4(a, b0, acc0);
    acc1 = wmma4(a, b1, acc1);
  }

  // Derived key ~ fold_in(SAMPLE_KEY=1234, t)
  unsigned dk0, dk1;
  threefry2x32(0u, 1234u, 0u, (unsigned)t, dk0, dk1);

#pragma unroll
  for (int m = 0; m < BATCH; ++m) {
    float comb = -INFINITY;
    unsigned vidx = 0xFFFFFFFFu;
    float mx = -INFINITY;
    if (val0) {
      const float lv = acc0[m] + b2[v0];
      logits[(size_t)m * VP + v0] = lv;
      unsigned r0, r1;
      threefry2x32(dk0, dk1, (unsigned)m, (unsigned)v0, r0, r1);
      float u = fmaxf((float)(r0 >> 8) * (1.0f / 16777216.0f), 1.1754944e-38f);
      const float g = -logf(fmaxf(-logf(u), 1.1754944e-38f));
      comb = lv + g; vidx = (unsigned)v0; mx = lv;
    }
    if (val1) {
      const float lv = acc1[m] + b2[v1];
      logits[(size_t)m * VP + v1] = lv;
      unsigned r0, r1;
      threefry2x32(dk0, dk1, (unsigned)m, (unsigned)v1, r0, r1);
      float u = fmaxf((float)(r0 >> 8) * (1.0f / 16777216.0f), 1.1754944e-38f);
      const float g = -logf(fmaxf(-logf(u), 1.1754944e-38f));
      const float cb = lv + g;
      if (cb > comb || (cb == comb && (unsigned)v1 < vidx)) { comb = cb; vidx = (unsigned)v1; }
      mx = fmaxf(mx, lv);
    }
    for (int off = 16; off > 0; off >>= 1) {
      const float oc = __shfl_xor(comb, off);
      const unsigned ov = __shfl_xor(vidx, off);
      if (oc > comb || (oc == comb && ov < vidx)) { comb = oc; vidx = ov; }
      mx = fmaxf(mx, __shfl_xor(mx, off));
    }
    if (lane == 0) {
      atomicMax(&red_max[m], ford(mx));
      const unsigned long long pk =
          ((unsigned long long)ford(comb) << 32) |
          (unsigned long long)(0xFFFFFFFFu - vidx);
      atomicMax(&red_arg[m], pk);
    }
  }
}

// Pass 2: sum(exp(logit - max)) per batch row.
__global__ void sumexp_kernel(const float* __restrict__ logits,
                              const unsigned* __restrict__ red_max,
                              float* red_sum) {
  const int v = blockIdx.x * blockDim.x + threadIdx.x;
  const int lane = threadIdx.x & 31;
#pragma unroll
  for (int m = 0; m < BATCH; ++m) {
    float val = 0.f;
    if (v < VOCAB) val = expf(logits[(size_t)m * VP + v] - fdec(red_max[m]));
    for (int off = 16; off > 0; off >>= 1) val += __shfl_xor(val, off);
    if (lane == 0) atomicAdd(&red_sum[m], val);
  }
}

// Finalize step t: sampled log-prob, token update, reset reduction slots.
__global__ void finalize_kernel(const float* __restrict__ logits,
                                unsigned* red_max, float* red_sum,
                                unsigned long long* red_arg,
                                const int* __restrict__ input_ids,
                                int* toks_ws, float* out_scores, int* out_tokens,
                                int t) {
  const int b = threadIdx.x;
  if (b < BATCH) {
    const unsigned long long pk = red_arg[b];
    const unsigned vidx = 0xFFFFFFFFu - (unsigned)(pk & 0xFFFFFFFFull);
    const float mx = fdec(red_max[b]);
    const float lse = mx + logf(red_sum[b]);
    const float slp = logits[(size_t)b * VP + vidx] - lse;
    const int cur = input_ids[b * TLEN + t];
    const int nxt = (cur == MASK_TOK) ? (int)vidx : cur;
    toks_ws[b * TLEN + t]    = nxt;
    out_tokens[b * TLEN + t] = nxt;
    out_scores[b * TLEN + t] = slp;
    // reset for next step
    red_max[b] = 0u;
    red_sum[b] = 0.f;
    red_arg[b] = 0ull;
  }
}

// ---------------- Host-side orchestration ----------------------------------

extern "C" void kernel_launch(void* const* d_in, const int* in_sizes, int n_in,
                              void* d_out, int out_size, void* d_ws, size_t ws_size,
                              hipStream_t stream) {
  const int*   input_ids = (const int*)d_in[0];
  /* d_in[1] attention_mask: unused by reference */
  const float* labels  = (const float*)d_in[2];
  const float* emb     = (const float*)d_in[3];
  const float* W_e2e   = (const float*)d_in[4];
  const float* b_e2e   = (const float*)d_in[5];
  const float* W_e2d   = (const float*)d_in[6];
  const float* b_e2d   = (const float*)d_in[7];
  const float* enc_Wih = (const float*)d_in[8];
  const float* enc_Whh = (const float*)d_in[9];
  const float* enc_bih = (const float*)d_in[10];
  const float* enc_bhh = (const float*)d_in[11];
  const float* dec_Wih = (const float*)d_in[12];
  const float* dec_Whh = (const float*)d_in[13];
  const float* dec_bih = (const float*)d_in[14];
  const float* dec_bhh = (const float*)d_in[15];
  const float* W_cls   = (const float*)d_in[16];
  const float* b_cls   = (const float*)d_in[17];
  const float* W_fc1   = (const float*)d_in[18];
  const float* b_fc1   = (const float*)d_in[19];
  const float* ln_g    = (const float*)d_in[20];
  const float* ln_b    = (const float*)d_in[21];
  const float* W_fc2   = (const float*)d_in[22];
  const float* b_fc2   = (const float*)d_in[23];
  (void)in_sizes; (void)n_in; (void)out_size; (void)ws_size;

  // Carve scratch from d_ws (256B-aligned slices).
  char* wp = (char*)d_ws;
  auto carve = [&](size_t bytes) {
    void* p = (void*)wp;
    wp += (bytes + 255) & ~(size_t)255;
    return p;
  };
  float* xe     = (float*)carve((size_t)TLEN * BATCH * HDIM * sizeof(float));
  float* xd     = (float*)carve((size_t)TLEN * BATCH * HDIM * sizeof(float));
  float* h_ws   = (float*)carve((size_t)NL * BATCH * HDIM * sizeof(float));
  float* c_ws   = (float*)carve((size_t)NL * BATCH * HDIM * sizeof(float));
  float* cls    = (float*)carve((size_t)BATCH * CDIM * sizeof(float));
  float* z1     = (float*)carve((size_t)BATCH * HC * sizeof(float));
  float* logits = (float*)carve((size_t)BATCH * VP * sizeof(float));
  int*   toks   = (int*)carve((size_t)BATCH * TLEN * sizeof(int));
  unsigned*           red_max = (unsigned*)carve(BATCH * sizeof(unsigned));
  float*              red_sum = (float*)carve(BATCH * sizeof(float));
  unsigned long long* red_arg = (unsigned long long*)carve(BATCH * sizeof(unsigned long long));

  float* out_scores = (float*)d_out;                       // [B,T] float32
  int*   out_tokens = ((int*)d_out) + BATCH * TLEN;        // [B,T] int32

  init_kernel<<<1, 64, 0, stream>>>(input_ids, toks, out_scores, out_tokens,
                                    red_max, red_sum, red_arg);
  cls_kernel<<<1, BATCH * CDIM, 0, stream>>>(labels, W_cls, b_cls, cls);
  enc_embed_kernel<<<(BATCH * TLEN) / 16, 256, 0, stream>>>(input_ids, emb, W_e2e,
                                                            b_e2e, xe);
  enc_lstm_kernel<<<1, 256, 0, stream>>>(xe, enc_Wih, enc_Whh, enc_bih, enc_bhh,
                                         h_ws, c_ws);

  for (int t = 1; t < TLEN; ++t) {
    dec_embed_kernel<<<1, 256, 0, stream>>>(toks, emb, W_e2d, b_e2d, xd, t);
    dec_step_kernel<<<1, 256, 0, stream>>>(xd, h_ws, c_ws, cls, dec_Wih, dec_Whh,
                                           dec_bih, dec_bhh, W_fc1, b_fc1, ln_g,
                                           ln_b, z1, t);
    fc2_kernel<<<VP / 128, 128, 0, stream>>>(z1, W_fc2, b_fc2, logits, red_max,
                                             red_arg, t);
    sumexp_kernel<<<(VOCAB + 255) / 256, 256, 0, stream>>>(logits, red_max, red_sum);
    finalize_kernel<<<1, 64, 0, stream>>>(logits, red_max, red_sum, red_arg,
                                          input_ids, toks, out_scores, out_tokens, t);
  }
}